// GMMAttention_89558658056840
// MI455X (gfx1250) — compile-verified
//
#include <hip/hip_runtime.h>
#include <math.h>

#define B_   64
#define T_   2000
#define E_   512
#define A_   128
#define R_   1024
#define K_   5
#define EPS_ 1e-5f
#define TCH  8
#define TCHLEN (T_ / TCH)   // 250

typedef float v2f __attribute__((ext_vector_type(2)));
typedef float v8f __attribute__((ext_vector_type(8)));

// ---------------------------------------------------------------------------
// Kernel 1: h = tanh(AHS @ W1 + b1) using V_WMMA_F32_16X16X4_F32.
// M=64 (batch), N=128 (A), K=1024 (R). One wave32 per 16x16 tile -> 4x8 tiles.
// ---------------------------------------------------------------------------
__global__ __launch_bounds__(32) void gemm1_wmma(const float* __restrict__ ahs,
                                                 const float* __restrict__ W1,
                                                 const float* __restrict__ b1,
                                                 float* __restrict__ h) {
  const int lane = threadIdx.x;       // 0..31, EXEC all-ones (no divergence)
  const int half = lane >> 4;         // 0 or 1
  const int l16  = lane & 15;
  const int m0   = blockIdx.y * 16;   // batch tile
  const int n0   = blockIdx.x * 16;   // A-dim tile

  v8f acc = {};
  const float* arow = ahs + (size_t)(m0 + l16) * R_; // A row for this lane's M
  const float* bcol = W1 + n0 + l16;                 // B column for this lane's N

  #pragma unroll 4
  for (int k0 = 0; k0 < R_; k0 += 4) {
    const int ka = k0 + 2 * half;
    // A-frag: lane holds A[m][ka], A[m][ka+1]  (ka even -> 8B aligned)
    v2f a = *(const v2f*)(arow + ka);
    // B-frag: lane holds B[ka][n], B[ka+1][n]
    v2f b;
    b.x = bcol[(size_t)ka * A_];
    b.y = bcol[(size_t)(ka + 1) * A_];
    acc = __builtin_amdgcn_wmma_f32_16x16x4_f32(
        /*neg_a=*/false, a, /*neg_b=*/false, b,
        /*c_mod=*/(short)0, acc, /*reuse_a=*/false, /*reuse_b=*/false);
  }

  const float bias = b1[n0 + l16];
  #pragma unroll
  for (int r = 0; r < 8; ++r) {
    const int m = m0 + r + 8 * half;       // C/D layout: VGPR r -> M=r (+8 for hi half)
    h[(size_t)m * A_ + n0 + l16] = tanhf(acc[r] + bias);
  }
}

// ---------------------------------------------------------------------------
// Kernel 2: params = h @ W2 + b2; mixture params per (b,k).
// One thread per batch row (64 threads).
// ---------------------------------------------------------------------------
__device__ __forceinline__ float softplusf(float x) {
  return (x > 0.0f) ? (x + log1pf(expf(-x))) : log1pf(expf(x));
}

__global__ __launch_bounds__(64) void mixture_params(const float* __restrict__ h,
                                                     const float* __restrict__ W2,
                                                     const float* __restrict__ b2,
                                                     const float* __restrict__ mu_prev,
                                                     float* __restrict__ coef,    // [B][K] omega/Z
                                                     float* __restrict__ mu_ws,   // [B][K]
                                                     float* __restrict__ inv2s2,  // [B][K] 0.5/sigma^2
                                                     float* __restrict__ out_mu)  // [B][K] output
{
  const int b = threadIdx.x;
  if (b >= B_) return;

  float p[3 * K_];
  #pragma unroll
  for (int j = 0; j < 3 * K_; ++j) p[j] = b2[j];

  const float* hb = h + (size_t)b * A_;
  for (int i = 0; i < A_; ++i) {
    const float hv = hb[i];
    const float* w = W2 + (size_t)i * (3 * K_);
    #pragma unroll
    for (int j = 0; j < 3 * K_; ++j) p[j] = fmaf(hv, w[j], p[j]);
  }

  // omega_hat = p[0..4], delta_hat = p[5..9], sigma_hat = p[10..14]
  float omax = p[0];
  #pragma unroll
  for (int k = 1; k < K_; ++k) omax = fmaxf(omax, p[k]);
  float oexp[K_], osum = 0.0f;
  #pragma unroll
  for (int k = 0; k < K_; ++k) { oexp[k] = expf(p[k] - omax); osum += oexp[k]; }
  const float oinv = 1.0f / osum;

  const float SQRT_2PI = 2.5066282746310002f;
  #pragma unroll
  for (int k = 0; k < K_; ++k) {
    const float omega = oexp[k] * oinv;
    const float delta = softplusf(p[K_ + k]);
    const float sigma = softplusf(p[2 * K_ + k]) + EPS_;
    const float Z     = SQRT_2PI * sigma;
    const float mu    = mu_prev[b * K_ + k] + delta;
    coef  [b * K_ + k] = omega / Z;
    mu_ws [b * K_ + k] = mu;
    inv2s2[b * K_ + k] = 0.5f / (sigma * sigma);
    out_mu[b * K_ + k] = mu;
  }
}

// ---------------------------------------------------------------------------
// Kernel 3: alignment (b,t) = sum_k coef*exp(-(t-mu)^2 * inv2s2), mask,
// softmax over T. One block per batch, 256 threads, shared-mem reductions.
// ---------------------------------------------------------------------------
__global__ __launch_bounds__(256) void align_softmax(const float* __restrict__ coef,
                                                     const float* __restrict__ mu_ws,
                                                     const float* __restrict__ inv2s2,
                                                     const unsigned char* __restrict__ mask,
                                                     float* __restrict__ weights) {
  const int b   = blockIdx.x;
  const int tid = threadIdx.x;
  __shared__ float red[256];
  __shared__ float sc[K_], sm[K_], si[K_];

  if (tid < K_) {
    sc[tid] = coef  [b * K_ + tid];
    sm[tid] = mu_ws [b * K_ + tid];
    si[tid] = inv2s2[b * K_ + tid];
  }
  __syncthreads();

  float vals[8];
  float lmax = -INFINITY;
  int n = 0;
  for (int t = tid; t < T_; t += 256, ++n) {
    float a = 0.0f;
    #pragma unroll
    for (int k = 0; k < K_; ++k) {
      const float d = (float)t - sm[k];
      a = fmaf(sc[k], expf(-d * d * si[k]), a);
    }
    if (mask[(size_t)b * T_ + t]) a = -INFINITY;
    vals[n] = a;
    lmax = fmaxf(lmax, a);
  }

  red[tid] = lmax;
  __syncthreads();
  for (int s = 128; s > 0; s >>= 1) {
    if (tid < s) red[tid] = fmaxf(red[tid], red[tid + s]);
    __syncthreads();
  }
  const float gmax = red[0];
  __syncthreads();

  float lsum = 0.0f;
  n = 0;
  for (int t = tid; t < T_; t += 256, ++n) {
    const float e = expf(vals[n] - gmax);
    vals[n] = e;
    lsum += e;
  }
  red[tid] = lsum;
  __syncthreads();
  for (int s = 128; s > 0; s >>= 1) {
    if (tid < s) red[tid] += red[tid + s];
    __syncthreads();
  }
  const float inv = 1.0f / red[0];

  n = 0;
  for (int t = tid; t < T_; t += 256, ++n)
    weights[(size_t)b * T_ + t] = vals[n] * inv;
}

// ---------------------------------------------------------------------------
// Kernel 4: context partials. The HBM-bound stage: streams 262 MB of memory.
// Grid (E/256, TCH, B): each block reduces a 250-step T-chunk for 256 e-cols.
// Coalesced along E; unrolled for memory-level parallelism.
// ---------------------------------------------------------------------------
__global__ __launch_bounds__(256) void ctx_partial(const float* __restrict__ weights,
                                                   const float* __restrict__ memory,
                                                   float* __restrict__ partial) {
  const int e = blockIdx.x * 256 + threadIdx.x; // 0..511
  const int c = blockIdx.y;                     // T chunk
  const int b = blockIdx.z;                     // batch

  const float* __restrict__ mb = memory + (size_t)b * T_ * E_ + e;
  const float* __restrict__ wb = weights + (size_t)b * T_;
  const int t0 = c * TCHLEN;
  const int t1 = t0 + TCHLEN;

  float acc = 0.0f;
  #pragma unroll 5
  for (int t = t0; t < t1; ++t)
    acc = fmaf(wb[t], mb[(size_t)t * E_], acc);

  partial[((size_t)c * B_ + b) * E_ + e] = acc;
}

// ---------------------------------------------------------------------------
// Kernel 5: deterministic reduction of the TCH partials into d_out context.
// ---------------------------------------------------------------------------
__global__ __launch_bounds__(256) void ctx_reduce(const float* __restrict__ partial,
                                                  float* __restrict__ ctx) {
  const int i = blockIdx.x * 256 + threadIdx.x; // 0 .. B*E-1
  float s = 0.0f;
  #pragma unroll
  for (int c = 0; c < TCH; ++c)
    s += partial[(size_t)c * B_ * E_ + i];
  ctx[i] = s;
}

// ---------------------------------------------------------------------------
extern "C" void kernel_launch(void* const* d_in, const int* in_sizes, int n_in,
                              void* d_out, int out_size, void* d_ws, size_t ws_size,
                              hipStream_t stream) {
  (void)in_sizes; (void)n_in; (void)out_size; (void)ws_size;

  const float*         ahs     = (const float*)d_in[0];         // (B,R)
  const float*         memory  = (const float*)d_in[1];         // (B,T,E)
  const unsigned char* mask    = (const unsigned char*)d_in[2]; // (B,T) bool
  const float*         mu_prev = (const float*)d_in[3];         // (B,K,1)
  const float*         W1      = (const float*)d_in[4];         // (R,A)
  const float*         b1      = (const float*)d_in[5];         // (A,)
  const float*         W2      = (const float*)d_in[6];         // (A,3K)
  const float*         b2      = (const float*)d_in[7];         // (3K,)

  // Output layout: [context B*E | weights B*T | mu B*K]
  float* out     = (float*)d_out;
  float* ctx     = out;
  float* weights = out + (size_t)B_ * E_;
  float* out_mu  = out + (size_t)B_ * E_ + (size_t)B_ * T_;

  // Workspace layout
  float* ws      = (float*)d_ws;
  float* h       = ws;                         // B*A   = 8192
  float* coef    = h + (size_t)B_ * A_;        // B*K   = 320
  float* mu_ws   = coef + B_ * K_;             // 320
  float* inv2s2  = mu_ws + B_ * K_;            // 320
  float* partial = inv2s2 + B_ * K_;           // TCH*B*E = 262144

  // 1) h = tanh(AHS @ W1 + b1)  [WMMA f32 16x16x4]
  gemm1_wmma<<<dim3(A_ / 16, B_ / 16), 32, 0, stream>>>(ahs, W1, b1, h);

  // 2) mixture parameters
  mixture_params<<<1, 64, 0, stream>>>(h, W2, b2, mu_prev,
                                       coef, mu_ws, inv2s2, out_mu);

  // 3) alignment + masked softmax over T -> attention weights
  align_softmax<<<B_, 256, 0, stream>>>(coef, mu_ws, inv2s2, mask, weights);

  // 4) context partials (HBM-bound stream over `memory`)
  ctx_partial<<<dim3(E_ / 256, TCH, B_), 256, 0, stream>>>(weights, memory, partial);

  // 5) deterministic partial reduction
  ctx_reduce<<<(B_ * E_) / 256, 256, 0, stream>>>(partial, ctx);
}